// Vad1Net_1047972020845
// MI455X (gfx1250) — compile-verified
//
#include <hip/hip_runtime.h>
#include <hip/hip_bf16.h>
#include <cmath>

// ---------------------------------------------------------------------------
// Vad1Net on MI455X (gfx1250, wave32).
// Kernel 1: a = tanh(x @ lc1_w^T + lc1_b)  via V_WMMA_F32_16X16X4_F32
// Kernel 2: GRU scan (16 batch rows / wave, h resident in LDS) with the
//           combined [a|h] @ [w_ih|w_hh]^T done as fp32 WMMA, gates + lc2
//           head fused. Only x, a, out touch HBM.
// ---------------------------------------------------------------------------

typedef __attribute__((ext_vector_type(2))) float v2f;
typedef __attribute__((ext_vector_type(8))) float v8f;

#define WMMA4F32(A, Bv, C)                                                   \
  __builtin_amdgcn_wmma_f32_16x16x4_f32(false, (A), false, (Bv), (short)0,   \
                                        (C), false, false)

static constexpr int Bsz  = 256;
static constexpr int Tlen = 2000;
static constexpr int Fin  = 42;   // lc1 input features
static constexpr int Hh   = 24;   // GRU hidden
static constexpr int Gg   = 72;   // 3*H gate columns

__device__ __forceinline__ float sigmoidf_(float v) {
  return 1.0f / (1.0f + __expf(-v));
}

// ---------------------------------------------------------------------------
// Kernel 1: per 16-row tile of the flattened [B*T, 42] input, compute
// a[16,24] = tanh(X[16,42] @ W1^T[42,24] + b1) with 11x2 fp32 WMMAs.
// A-frag layout (16x4 f32): lane l<16 holds (M=l, K=kc..kc+1) in v0/v1,
// lane l+16 holds (M=l, K=kc+2..kc+3).  B-frag (4x16): lane n<16 holds
// (K=kc, N=n)/(K=kc+1, N=n); lanes 16..31 hold K=kc+2/kc+3.
// C/D layout: vgpr j <-> (M = j + 8*(lane>=16), N = lane&15).
// All bounds handling is done by unconditional in-range loads + value
// selects (v_cndmask), never by divergent guarded loads.
// ---------------------------------------------------------------------------
__global__ __launch_bounds__(32) void lc1_tanh_kernel(
    const float* __restrict__ x, const float* __restrict__ w1,
    const float* __restrict__ b1, float* __restrict__ aOut) {
  const int  lane = threadIdx.x;
  const int  nl   = lane & 15;
  const int  ko   = (lane >> 4) * 2;     // K offset for this half-wave
  const int  mo   = (lane >> 4) * 8;     // M offset for C layout
  const bool hiHalf = (lane >= 16);
  const v2f  zero2 = {0.0f, 0.0f};

  // --- B fragments for W1^T: K padded 42->44, N padded 24->32, in regs ----
  v2f B0[11], B1[11];
  const int  n1  = 16 + nl;
  const bool n1v = (n1 < Hh);
  const int  n1c = n1v ? n1 : (Hh - 1);  // clamped row
#pragma unroll
  for (int kc = 0; kc < 11; ++kc) {
    const int k   = kc * 4 + ko;
    const int kc0 = (k     < Fin) ? k     : 0;  // clamped cols
    const int kc1 = (k + 1 < Fin) ? k + 1 : 0;
    const float w00 = w1[nl  * Fin + kc0];
    const float w01 = w1[nl  * Fin + kc1];
    const float w10 = w1[n1c * Fin + kc0];
    const float w11 = w1[n1c * Fin + kc1];
    B0[kc].x = (k     < Fin) ? w00 : 0.0f;
    B0[kc].y = (k + 1 < Fin) ? w01 : 0.0f;
    B1[kc].x = (n1v && (k     < Fin)) ? w10 : 0.0f;
    B1[kc].y = (n1v && (k + 1 < Fin)) ? w11 : 0.0f;
  }
  const float bA = b1[nl];
  const float bB = n1v ? b1[n1c] : 0.0f;

  const int NT = (Bsz * Tlen) / 16;  // 32000 tiles, exact
  for (int tile = blockIdx.x; tile < NT; tile += gridDim.x) {
    const float* xr = x + (size_t)(tile * 16 + nl) * Fin;

    // Prefetch next tile's rows while we compute this one.
    if (tile + (int)gridDim.x < NT)
      __builtin_prefetch(xr + (size_t)gridDim.x * 16 * Fin, 0, 1);

    // Dual accumulators per N-tile: break the WMMA->WMMA RAW chain.
    v8f a0e = {0, 0, 0, 0, 0, 0, 0, 0}, a0o = {0, 0, 0, 0, 0, 0, 0, 0};
    v8f a1e = {0, 0, 0, 0, 0, 0, 0, 0}, a1o = {0, 0, 0, 0, 0, 0, 0, 0};
    // Chunks 0..9: K = kc*4+ko+1 <= 39 < 42, always in range -> plain v2f
    // loads (8B aligned: row stride 42 floats is even, k is even).
#pragma unroll
    for (int kc = 0; kc < 10; kc += 2) {
      const v2f Ae = *(const v2f*)(xr + kc * 4 + ko);
      const v2f Ao = *(const v2f*)(xr + (kc + 1) * 4 + ko);
      a0e = WMMA4F32(Ae, B0[kc], a0e);
      a1e = WMMA4F32(Ae, B1[kc], a1e);
      a0o = WMMA4F32(Ao, B0[kc + 1], a0o);
      a1o = WMMA4F32(Ao, B1[kc + 1], a1o);
    }
    // Tail chunk kc=10: lanes<16 need K=40..41 (in range); lanes>=16 need
    // K=42..43 (zero pad). Unconditional in-range load + value select.
    {
      const v2f t = *(const v2f*)(xr + 40);
      const v2f A = hiHalf ? zero2 : t;
      a0e = WMMA4F32(A, B0[10], a0e);
      a1e = WMMA4F32(A, B1[10], a1e);
    }

    float* arow = aOut + (size_t)tile * 16 * Hh;
#pragma unroll
    for (int j = 0; j < 8; ++j)
      arow[(j + mo) * Hh + nl] = tanhf(a0e[j] + a0o[j] + bA);
    if (nl < 8) {
#pragma unroll
      for (int j = 0; j < 8; ++j)
        arow[(j + mo) * Hh + 16 + nl] = tanhf(a1e[j] + a1o[j] + bB);
    }
  }
}

// ---------------------------------------------------------------------------
// Kernel 2: GRU scan + lc2 head. One wave per 16 batch rows; h lives in LDS.
// Per step:  preacts[16,72] = [a_t | h] @ [w_ih | w_hh]^T  (K=48, N=80 padded)
// r,z columns (g<48) accumulate all K; n columns keep x-side and h-side
// accumulators separate (needed for  n = tanh(xn + r*hn)).
// Each accumulator is split even/odd over K-chunks to halve the serial WMMA
// dependency chain on the scan's critical path.
// ---------------------------------------------------------------------------
__global__ __launch_bounds__(32) void gru_scan_kernel(
    const float* __restrict__ a, const float* __restrict__ w_ih,
    const float* __restrict__ w_hh, const float* __restrict__ b_ih,
    const float* __restrict__ b_hh, const float* __restrict__ lc2_w,
    const float* __restrict__ lc2_b, float* __restrict__ out) {
  __shared__ __align__(16) float Wlds[80][48];  // [n][k] combined weights
  __shared__ __align__(16) float hS[16][24];    // hidden state (m, c)
  __shared__ __align__(16) float pS[16][96];    // 0..47 rz, 48..71 xn, 72..95 hn
  __shared__ float bsum[48], bihn[24], bhhn[24], lc2S[24];

  const int lane = threadIdx.x;
  const int nl   = lane & 15;
  const int ko   = (lane >> 4) * 2;
  const int mo   = (lane >> 4) * 8;
  const int b0   = blockIdx.x * 16;

  // Combined weight image: k<24 -> w_ih[n][k], else w_hh[n][k-24]; n>=72 -> 0
  for (int idx = lane; idx < 80 * 48; idx += 32) {
    const int n = idx / 48, k = idx % 48;
    const int nc = (n < Gg) ? n : (Gg - 1);
    const float v = (k < Hh) ? w_ih[nc * Hh + k] : w_hh[nc * Hh + (k - Hh)];
    Wlds[n][k] = (n < Gg) ? v : 0.0f;
  }
  if (lane < 24) {
    bihn[lane] = b_ih[48 + lane];
    bhhn[lane] = b_hh[48 + lane];
    lc2S[lane] = lc2_w[lane];
  }
  for (int g = lane; g < 48; g += 32) bsum[g] = b_ih[g] + b_hh[g];

  const int gm = lane >> 1;          // gate-math row (batch m)
  const int cb = (lane & 1) * 12;    // gate-math column base
#pragma unroll
  for (int i = 0; i < 12; ++i) hS[gm][cb + i] = 0.0f;  // h0 = 0
  const float l2b = lc2_b[0];
  __syncthreads();

  for (int t = 0; t < Tlen; ++t) {
    // ---- A fragments: K 0..23 from a(global), K 24..47 from hS(LDS) ----
    v2f Af[12];
    const float* ab = a + ((size_t)(b0 + nl) * Tlen + t) * Hh;
#pragma unroll
    for (int kc = 0; kc < 6; ++kc)
      Af[kc] = *(const v2f*)(ab + kc * 4 + ko);
#pragma unroll
    for (int kc = 6; kc < 12; ++kc)
      Af[kc] = *(const v2f*)(&hS[nl][(kc - 6) * 4 + ko]);

    // Prefetch next timestep's a-row into the cache hierarchy.
    if (t + 1 < Tlen) __builtin_prefetch(ab + Hh, 0, 1);

    // ---- r,z gate columns (N-tiles 0..2): full-K accumulation ----
#pragma unroll
    for (int nt = 0; nt < 3; ++nt) {
      v8f ae = {0, 0, 0, 0, 0, 0, 0, 0};
      v8f ao = {0, 0, 0, 0, 0, 0, 0, 0};
#pragma unroll
      for (int kc = 0; kc < 12; kc += 2) {
        const v2f bve = *(const v2f*)(&Wlds[nt * 16 + nl][kc * 4 + ko]);
        const v2f bvo = *(const v2f*)(&Wlds[nt * 16 + nl][(kc + 1) * 4 + ko]);
        ae = WMMA4F32(Af[kc], bve, ae);
        ao = WMMA4F32(Af[kc + 1], bvo, ao);
      }
#pragma unroll
      for (int j = 0; j < 8; ++j) pS[j + mo][nt * 16 + nl] = ae[j] + ao[j];
    }
    // ---- n gate columns (N-tiles 3..4): split x-side / h-side ----
#pragma unroll
    for (int nt = 3; nt < 5; ++nt) {
      v8f aXe = {0, 0, 0, 0, 0, 0, 0, 0}, aXo = {0, 0, 0, 0, 0, 0, 0, 0};
      v8f aHe = {0, 0, 0, 0, 0, 0, 0, 0}, aHo = {0, 0, 0, 0, 0, 0, 0, 0};
#pragma unroll
      for (int kc = 0; kc < 6; kc += 2) {
        const v2f bve = *(const v2f*)(&Wlds[nt * 16 + nl][kc * 4 + ko]);
        const v2f bvo = *(const v2f*)(&Wlds[nt * 16 + nl][(kc + 1) * 4 + ko]);
        aXe = WMMA4F32(Af[kc], bve, aXe);
        aXo = WMMA4F32(Af[kc + 1], bvo, aXo);
      }
#pragma unroll
      for (int kc = 6; kc < 12; kc += 2) {
        const v2f bve = *(const v2f*)(&Wlds[nt * 16 + nl][kc * 4 + ko]);
        const v2f bvo = *(const v2f*)(&Wlds[nt * 16 + nl][(kc + 1) * 4 + ko]);
        aHe = WMMA4F32(Af[kc], bve, aHe);
        aHo = WMMA4F32(Af[kc + 1], bvo, aHo);
      }
      const int c = (nt - 3) * 16 + nl;   // n-gate column 0..31 (valid < 24)
      if (c < Hh) {
#pragma unroll
        for (int j = 0; j < 8; ++j) {
          pS[j + mo][48 + c] = aXe[j] + aXo[j];
          pS[j + mo][72 + c] = aHe[j] + aHo[j];
        }
      }
    }
    __syncthreads();

    // ---- gates, state update, fused lc2 head ----
    float s = 0.0f;
#pragma unroll
    for (int i = 0; i < 12; ++i) {
      const int c = cb + i;
      const float r  = sigmoidf_(pS[gm][c] + bsum[c]);
      const float z  = sigmoidf_(pS[gm][24 + c] + bsum[24 + c]);
      const float xn = pS[gm][48 + c] + bihn[c];
      const float hn = pS[gm][72 + c] + bhhn[c];
      const float nn = tanhf(xn + r * hn);
      const float hv = (1.0f - z) * nn + z * hS[gm][c];
      hS[gm][c] = hv;
      s += lc2S[c] * hv;
    }
    s += __shfl_xor(s, 1, 32);
    if ((lane & 1) == 0)
      out[(size_t)(b0 + gm) * Tlen + t] = sigmoidf_(s + l2b);
    __syncthreads();
  }
}

// ---------------------------------------------------------------------------
// Host launcher. d_ws holds the intermediate a = tanh(lc1(x)): B*T*H*4 =
// 49,152,000 bytes.
// ---------------------------------------------------------------------------
extern "C" void kernel_launch(void* const* d_in, const int* in_sizes, int n_in,
                              void* d_out, int out_size, void* d_ws,
                              size_t ws_size, hipStream_t stream) {
  (void)in_sizes; (void)n_in; (void)out_size; (void)ws_size;
  const float* x     = (const float*)d_in[0];
  const float* lc1_w = (const float*)d_in[1];
  const float* lc1_b = (const float*)d_in[2];
  const float* w_ih  = (const float*)d_in[3];
  const float* w_hh  = (const float*)d_in[4];
  const float* b_ih  = (const float*)d_in[5];
  const float* b_hh  = (const float*)d_in[6];
  const float* lc2_w = (const float*)d_in[7];
  const float* lc2_b = (const float*)d_in[8];
  float* out  = (float*)d_out;
  float* aBuf = (float*)d_ws;  // [B, T, H] fp32

  lc1_tanh_kernel<<<2048, 32, 0, stream>>>(x, lc1_w, lc1_b, aBuf);
  gru_scan_kernel<<<Bsz / 16, 32, 0, stream>>>(aBuf, w_ih, w_hh, b_ih, b_hh,
                                               lc2_w, lc2_b, out);
}